// SymmetryNet_14551349199338
// MI455X (gfx1250) — compile-verified
//
#include <hip/hip_runtime.h>
#include <math.h>

// ---------------------------------------------------------------------------
// SymmetryNet pipeline for gfx1250 (MI455X), fp32 WMMA path.
// B=2, N=512, K=16 neighbors, MLP 3->32->64, 24 octahedral rotation diagonals,
// soft-top-k (k=2, tau=1e-2), plane fit output [B, N, 4].
// ---------------------------------------------------------------------------

#define BB   2
#define NN   512
#define KNN  16
#define C1   32
#define C2   64
#define NR   24
#define R2   (0.2f * 0.2f)
#define INV_TAU 100.0f

typedef __attribute__((ext_vector_type(2))) float v2f;
typedef __attribute__((ext_vector_type(8))) float v8f;

// diag(R) for the 24 proper signed-permutation rotations. Order is irrelevant
// (only a mean over r is taken). Non-fixed-point permutations give zero diag.
__constant__ float g_diag[NR][3] = {
    { 1, 1, 1}, { 1,-1,-1}, {-1, 1,-1}, {-1,-1, 1},   // identity perm
    { 1, 0, 0}, { 1, 0, 0}, {-1, 0, 0}, {-1, 0, 0},   // perm (0,2,1)
    { 0, 0,-1}, { 0, 0, 1}, { 0, 0, 1}, { 0, 0,-1},   // perm (1,0,2)
    { 0, 0, 0}, { 0, 0, 0}, { 0, 0, 0}, { 0, 0, 0},   // perm (1,2,0)
    { 0, 0, 0}, { 0, 0, 0}, { 0, 0, 0}, { 0, 0, 0},   // perm (2,0,1)
    { 0, 1, 0}, { 0,-1, 0}, { 0, 1, 0}, { 0,-1, 0}    // perm (2,1,0)
};

// ---------------------------------------------------------------------------
// Kernel 1: ball-query kNN. One wave per query point. 16 sequential argmin
// passes over an LDS distance array (intra-wave only; LDS is in-order per wave).
// ---------------------------------------------------------------------------
#define KNN_WAVES 8
__global__ void __launch_bounds__(256) knn_kernel(const float* __restrict__ pos,
                                                  float* __restrict__ gp) {
  __shared__ float d2s[KNN_WAVES][NN];
  const int lane = threadIdx.x & 31;
  const int wav  = threadIdx.x >> 5;
  const int p    = blockIdx.x * KNN_WAVES + wav;       // 0..B*N-1
  const int b    = p >> 9;
  const int n    = p & (NN - 1);
  const float* pb = pos + b * NN * 3;
  const float cx = pb[n * 3 + 0], cy = pb[n * 3 + 1], cz = pb[n * 3 + 2];

  for (int j = lane; j < NN; j += 32) {
    float dx = pb[j * 3 + 0] - cx;
    float dy = pb[j * 3 + 1] - cy;
    float dz = pb[j * 3 + 2] - cz;
    d2s[wav][j] = dx * dx + dy * dy + dz * dz;
  }
  asm volatile("" ::: "memory");

  int nearest = 0;
  for (int it = 0; it < KNN; ++it) {
    float best = 3.4e38f;
    int   bi   = 0x7fffffff;
    for (int j = lane; j < NN; j += 32) {
      float v = d2s[wav][j];
      if (v < best || (v == best && j < bi)) { best = v; bi = j; }
    }
    // wave32 butterfly argmin (ties -> lowest index, matching lax.top_k)
    for (int off = 16; off >= 1; off >>= 1) {
      float ov = __shfl_xor(best, off, 32);
      int   oi = __shfl_xor(bi, off, 32);
      if (ov < best || (ov == best && oi < bi)) { best = ov; bi = oi; }
    }
    if (it == 0) nearest = bi;
    const int use = (best <= R2) ? bi : nearest;  // outside radius -> nearest (self)
    if (lane == 0) {
      d2s[wav][bi] = 3.4e38f;
      float* o = gp + (size_t)(p * KNN + it) * 3;
      o[0] = pb[use * 3 + 0] - cx;
      o[1] = pb[use * 3 + 1] - cy;
      o[2] = pb[use * 3 + 2] - cz;
    }
    asm volatile("" ::: "memory");
  }
}

// ---------------------------------------------------------------------------
// Kernel 2: rotation MLP via fp32 WMMA 16x16x4.
// Block = 256 threads = 8 waves; one block per (b,n); each wave does 3 of the
// 24 rotations. Tile: M=16 neighbors, K=channels, N=output channels.
//   layer1: (16x4) x (4x16) x 2 tiles -> h1 16x32  (coords padded K=4)
//   layer2: (16x32) x (32x16) x 4 tiles, 8 chained K=4 WMMAs each
// Fragment layouts per CDNA5 ISA 7.12.2:
//   A (16x4 f32): lanes 0-15 -> M=lane, V0=K0,V1=K1 ; lanes 16-31 -> K2,K3
//   B (4x16 f32): lanes 0-15 -> N=lane, V0=K0,V1=K1 ; lanes 16-31 -> K2,K3
//   C/D (16x16 f32): VGPR v -> row v (lanes 0-15) / v+8 (lanes 16-31), col=lane&15
// ---------------------------------------------------------------------------
__global__ void __launch_bounds__(256) mlp_kernel(const float* __restrict__ gp,
                                                  const float* __restrict__ W1,
                                                  const float* __restrict__ b1,
                                                  const float* __restrict__ W2,
                                                  const float* __restrict__ b2,
                                                  float* __restrict__ fOut) {
  __shared__ float sW1[C1 * 4];        // W1 padded K: sW1[c*4 + k], k=3 -> 0
  __shared__ float sB1[C1];
  __shared__ float sW2[C2 * C1];       // sW2[o*32 + c]
  __shared__ float sB2[C2];
  __shared__ float h1lds[8][16 * C1];  // per-wave h1 tile [m][c]
  __shared__ float fsum[C2];

  const int tid  = threadIdx.x;
  const int lane = tid & 31;
  const int wav  = tid >> 5;
  const int half = lane >> 4;
  const int mcol = lane & 15;
  const int p = blockIdx.x;            // (b,n) flat

  for (int i = tid; i < C1 * 3; i += 256) sW1[(i / 3) * 4 + (i % 3)] = W1[i];
  if (tid < C1) { sW1[tid * 4 + 3] = 0.0f; sB1[tid] = b1[tid]; }
  for (int i = tid; i < C2 * C1; i += 256) sW2[i] = W2[i];
  if (tid < C2) { sB2[tid] = b2[tid]; fsum[tid] = 0.0f; }
  __syncthreads();

  // this lane's neighbor row (halves duplicate M=0..15)
  const float* g = gp + (size_t)p * KNN * 3;
  const float gx = g[mcol * 3 + 0];
  const float gy = g[mcol * 3 + 1];
  const float gz = g[mcol * 3 + 2];

  float accT[4] = {0.f, 0.f, 0.f, 0.f};  // per-lane channel sums over rotations

  for (int it = 0; it < 3; ++it) {
    const int r = wav + 8 * it;
    const float dg0 = g_diag[r][0], dg1 = g_diag[r][1], dg2 = g_diag[r][2];

    // ---- layer 1: A = neighbors*diag (16x4) ----
    v2f a;
    if (half == 0) { a[0] = gx * dg0; a[1] = gy * dg1; }
    else           { a[0] = gz * dg2; a[1] = 0.0f; }

    for (int t = 0; t < 2; ++t) {
      const int c = t * 16 + mcol;     // output channel = B column
      v2f bw;
      bw[0] = sW1[c * 4 + 2 * half + 0];
      bw[1] = sW1[c * 4 + 2 * half + 1];
      v8f cz = {};
      v8f dd = __builtin_amdgcn_wmma_f32_16x16x4_f32(false, a, false, bw,
                                                     (short)0, cz, false, false);
      const float bias = sB1[c];
      // bias + relu, store h1 in [m][c] order for layer-2 A fragments
      for (int v = 0; v < 8; ++v) {
        float x = dd[v] + bias;
        x = x > 0.f ? x : 0.f;
        h1lds[wav][(v + 8 * half) * C1 + c] = x;
      }
    }
    __syncthreads();   // intra-wave transpose through LDS (waves are aligned)

    // ---- layer 2: h1 (16x32) x W2^T (32x64) as 4 tiles x 8 chained K=4 ----
    for (int t = 0; t < 4; ++t) {
      const int c = t * 16 + mcol;
      v8f accv = {};
      for (int kk = 0; kk < 8; ++kk) {
        const int kb = 4 * kk + 2 * half;
        v2f a2, bw2;
        a2[0]  = h1lds[wav][mcol * C1 + kb + 0];
        a2[1]  = h1lds[wav][mcol * C1 + kb + 1];
        bw2[0] = sW2[c * C1 + kb + 0];
        bw2[1] = sW2[c * C1 + kb + 1];
        accv = __builtin_amdgcn_wmma_f32_16x16x4_f32(false, a2, false, bw2,
                                                     (short)0, accv, false, false);
      }
      // bias + relu + max over the 16 neighbor rows (8 VGPRs x 2 lane-halves)
      const float bias = sB2[c];
      float colmax = -3.4e38f;
      for (int v = 0; v < 8; ++v) {
        float x = accv[v] + bias;
        x = x > 0.f ? x : 0.f;
        colmax = fmaxf(colmax, x);
      }
      colmax = fmaxf(colmax, __shfl_xor(colmax, 16, 32));
      accT[t] += colmax;
    }
    __syncthreads();
  }

  // combine the 8 waves (3 rotations each) -> mean over 24
  if (half == 0) {
    atomicAdd(&fsum[ 0 + mcol], accT[0]);
    atomicAdd(&fsum[16 + mcol], accT[1]);
    atomicAdd(&fsum[32 + mcol], accT[2]);
    atomicAdd(&fsum[48 + mcol], accT[3]);
  }
  __syncthreads();
  if (tid < C2) fOut[(size_t)p * C2 + tid] = fsum[tid] * (1.0f / 24.0f);
}

// ---------------------------------------------------------------------------
// Kernel 3: p=0.5-norm pairwise distances, 2-step soft-top-k, plane fit.
// One block (256 threads) per (b,i) row; each thread owns 2 columns j.
// ---------------------------------------------------------------------------
__device__ __forceinline__ float blockMax(float v, float* red) {
  const int lane = threadIdx.x & 31, wav = threadIdx.x >> 5;
  for (int off = 16; off >= 1; off >>= 1) v = fmaxf(v, __shfl_xor(v, off, 32));
  __syncthreads();
  if (lane == 0) red[wav] = v;
  __syncthreads();
  float r = red[0];
  for (int w = 1; w < 8; ++w) r = fmaxf(r, red[w]);
  return r;
}
__device__ __forceinline__ float blockSum(float v, float* red) {
  const int lane = threadIdx.x & 31, wav = threadIdx.x >> 5;
  for (int off = 16; off >= 1; off >>= 1) v += __shfl_xor(v, off, 32);
  __syncthreads();
  if (lane == 0) red[wav] = v;
  __syncthreads();
  float r = red[0];
  for (int w = 1; w < 8; ++w) r += red[w];
  return r;
}

__global__ void __launch_bounds__(256) plane_kernel(const float* __restrict__ f,
                                                    const float* __restrict__ pos,
                                                    float* __restrict__ out) {
  __shared__ float fi[C2];
  __shared__ float red[8];
  const int tid = threadIdx.x;
  const int p = blockIdx.x;
  const int b = p >> 9;
  const int i = p & (NN - 1);
  const float* fb = f + (size_t)b * NN * C2;
  const float* pb = pos + (size_t)b * NN * 3;

  if (tid < C2) fi[tid] = fb[(size_t)i * C2 + tid];
  __syncthreads();

  // distances / logits for this thread's two columns
  float s0, s1;
  {
    const float* fj = fb + (size_t)tid * C2;
    float acc = 0.f;
    for (int c = 0; c < C2; ++c) acc += sqrtf(fabsf(fi[c] - fj[c]) + 1e-8f);
    s0 = -(acc * acc) * INV_TAU;
  }
  {
    const float* fj = fb + (size_t)(tid + 256) * C2;
    float acc = 0.f;
    for (int c = 0; c < C2; ++c) acc += sqrtf(fabsf(fi[c] - fj[c]) + 1e-8f);
    s1 = -(acc * acc) * INV_TAU;
  }

  // softmax pass 1
  const float m1 = blockMax(fmaxf(s0, s1), red);
  float e0 = __expf(s0 - m1), e1 = __expf(s1 - m1);
  const float sum1 = blockSum(e0 + e1, red);
  const float w00 = e0 / sum1, w01 = e1 / sum1;

  // mask = log1p(-clip(w, 0, 1-1e-6)); softmax pass 2
  const float s20 = s0 + log1pf(-fminf(w00, 1.f - 1e-6f));
  const float s21 = s1 + log1pf(-fminf(w01, 1.f - 1e-6f));
  const float m2 = blockMax(fmaxf(s20, s21), red);
  float g0 = __expf(s20 - m2), g1 = __expf(s21 - m2);
  const float sum2 = blockSum(g0 + g1, red);
  const float w10 = g0 / sum2, w11 = g1 / sum2;

  // pos_prime = corr @ pos
  float px = w10 * pb[tid * 3 + 0] + w11 * pb[(tid + 256) * 3 + 0];
  float py = w10 * pb[tid * 3 + 1] + w11 * pb[(tid + 256) * 3 + 1];
  float pz = w10 * pb[tid * 3 + 2] + w11 * pb[(tid + 256) * 3 + 2];
  px = blockSum(px, red);
  py = blockSum(py, red);
  pz = blockSum(pz, red);

  if (tid == 0) {
    const float rx = pb[i * 3 + 0], ry = pb[i * 3 + 1], rz = pb[i * 3 + 2];
    const float vx = px - rx, vy = py - ry, vz = pz - rz;
    float nrm = sqrtf(vx * vx + vy * vy + vz * vz);
    nrm = fmaxf(nrm, 1e-12f);
    const float nx = vx / nrm, ny = vy / nrm, nz = vz / nrm;
    const float fx = 0.5f * (rx + px), fy = 0.5f * (ry + py), fz = 0.5f * (rz + pz);
    const float dpl = -(nx * fx + ny * fy + nz * fz);
    float* o = out + (size_t)p * 4;
    o[0] = nx; o[1] = ny; o[2] = nz; o[3] = dpl;
  }
}

// ---------------------------------------------------------------------------
extern "C" void kernel_launch(void* const* d_in, const int* in_sizes, int n_in,
                              void* d_out, int out_size, void* d_ws, size_t ws_size,
                              hipStream_t stream) {
  (void)in_sizes; (void)n_in; (void)out_size; (void)ws_size;
  const float* pos = (const float*)d_in[0];
  const float* W1  = (const float*)d_in[1];
  const float* b1  = (const float*)d_in[2];
  const float* W2  = (const float*)d_in[3];
  const float* b2  = (const float*)d_in[4];
  float* out = (float*)d_out;

  float* gp = (float*)d_ws;                       // [B*N, 16, 3]
  float* f  = gp + (size_t)BB * NN * KNN * 3;     // [B*N, 64]

  knn_kernel  <<<(BB * NN) / KNN_WAVES, 256, 0, stream>>>(pos, gp);
  mlp_kernel  <<<BB * NN, 256, 0, stream>>>(gp, W1, b1, W2, b2, f);
  plane_kernel<<<BB * NN, 256, 0, stream>>>(f, pos, out);
}